// HyperGNNBinaryComposeLayer_15951508537733
// MI455X (gfx1250) — compile-verified
//
#include <hip/hip_runtime.h>
#include <hip/hip_bf16.h>

typedef __attribute__((ext_vector_type(2))) float v2f;
typedef __attribute__((ext_vector_type(8))) float v8f;

#define BS 2
#define NS 48
#define NO 48
#define REL_DIM 512
#define MEM_DIM 256
#define ROWS_P (BS * NS * NO)      // 4608 rows for p1/p2
#define LSTRIDE 260                // padded LDS row stride (floats); 260 = 4*65 keeps 16B align

// ---------------------------------------------------------------------------
// Small out-of-place transpose: out[c*R + r] = in[r*C + c]
// ---------------------------------------------------------------------------
__global__ void transpose_kernel(const float* __restrict__ in, float* __restrict__ out,
                                 int R, int C) {
  int i = blockIdx.x * blockDim.x + threadIdx.x;
  if (i < R * C) {
    int r = i / C, c = i % C;
    out[(size_t)c * R + r] = in[i];
  }
}

// ---------------------------------------------------------------------------
// p = rel @ W + b   via V_WMMA_F32_16X16X4_F32.
// Grid: (ROWS_P/16, 2). blockIdx.y selects (W1,b1,p1) vs (W2,b2,p2).
// 512 threads = 16 waves; wave w owns output columns [16w, 16w+16).
// ---------------------------------------------------------------------------
__global__ void proj_kernel(const float* __restrict__ rel,
                            const float* __restrict__ W1T, const float* __restrict__ b1,
                            const float* __restrict__ W2T, const float* __restrict__ b2,
                            float* __restrict__ p1, float* __restrict__ p2) {
  const float* WT   = blockIdx.y ? W2T : W1T;
  const float* bias = blockIdx.y ? b2  : b1;
  float*       outp = blockIdx.y ? p2  : p1;

  const int tid   = threadIdx.x;
  const int wave  = tid >> 5;
  const int lane  = tid & 31;
  const int laneN = lane & 15;
  const int kOff  = (lane < 16) ? 0 : 2;   // A/B fragment K sub-offset per half-wave
  const int n0    = wave * 16;
  const int rowBase = blockIdx.x * 16;

  const float2* aP = reinterpret_cast<const float2*>(
      rel + (size_t)(rowBase + laneN) * REL_DIM + kOff);
  const float2* bP = reinterpret_cast<const float2*>(
      WT + (size_t)(n0 + laneN) * REL_DIM + kOff);

  v8f c = {};
#pragma unroll 4
  for (int kk = 0; kk < REL_DIM; kk += 4) {
    float2 af = aP[kk >> 1];
    float2 bf = bP[kk >> 1];
    v2f a; a.x = af.x; a.y = af.y;
    v2f b; b.x = bf.x; b.y = bf.y;
    c = __builtin_amdgcn_wmma_f32_16x16x4_f32(false, a, false, b, (short)0, c, false, false);
  }

  const int rOff = (lane < 16) ? 0 : 8;     // C/D layout: VGPR i -> row i or i+8
  const float bv = bias[n0 + laneN];
#pragma unroll
  for (int i = 0; i < 8; ++i)
    outp[(size_t)(rowBase + rOff + i) * MEM_DIM + n0 + laneN] = c[i] + bv;
}

// ---------------------------------------------------------------------------
// Encoder + LayerNorm. One block per (b,s,o1) = 4608 blocks, 512 threads.
// prodT[48][LSTRIDE] staged once; all 3 M-tiles accumulated in ONE k-pass
// (3 accumulators share each B fragment); h[48][LSTRIDE] written, then LN.
// ---------------------------------------------------------------------------
__global__ void encode_kernel(const float* __restrict__ p1, const float* __restrict__ p2,
                              const float* __restrict__ WeT, const float* __restrict__ be,
                              const float* __restrict__ gamma, const float* __restrict__ beta,
                              float* __restrict__ out) {
  extern __shared__ float smem[];
  float* prodT = smem;                   // 48 * LSTRIDE floats
  float* hT    = smem + NO * LSTRIDE;    // 48 * LSTRIDE floats

  const int bso1   = blockIdx.x;               // (b*NS+s)*NO + o1
  const int base48 = (bso1 / NO) * NO;         // p2 row base for this (b,s)
  const int tid    = threadIdx.x;

  // ---- stage prod[o2][k] = p1[bso1][k] * p2[base48+o2][k] (float4 granularity) ----
  {
    const int k4 = tid & 63;                   // float4 index within a row
    const float4 p1v = *reinterpret_cast<const float4*>(p1 + (size_t)bso1 * MEM_DIM + 4 * k4);
#pragma unroll
    for (int j = 0; j < 6; ++j) {
      const int idx = tid + j * 512;           // over 48*64 float4s
      const int o2  = idx >> 6;
      const float4 pv = *reinterpret_cast<const float4*>(
          p2 + (size_t)(base48 + o2) * MEM_DIM + 4 * k4);
      float4 pr;
      pr.x = p1v.x * pv.x; pr.y = p1v.y * pv.y;
      pr.z = p1v.z * pv.z; pr.w = p1v.w * pv.w;
      *reinterpret_cast<float4*>(&prodT[o2 * LSTRIDE + 4 * k4]) = pr;
    }
  }
  __syncthreads();

  const int wave  = tid >> 5;
  const int lane  = tid & 31;
  const int laneN = lane & 15;
  const int kOff  = (lane < 16) ? 0 : 2;
  const int n0    = wave * 16;
  const int rOff  = (lane < 16) ? 0 : 8;

  const float2* bP = reinterpret_cast<const float2*>(
      WeT + (size_t)(n0 + laneN) * MEM_DIM + kOff);
  const float2* aP = reinterpret_cast<const float2*>(
      &prodT[laneN * LSTRIDE + kOff]);
  const int rowStride2 = 8 * LSTRIDE;          // 16 rows in float2 units

  v8f c0 = {}, c1 = {}, c2 = {};
#pragma unroll 4
  for (int kk = 0; kk < MEM_DIM; kk += 4) {
    float2 bf = bP[kk >> 1];                   // shared by all 3 M-tiles
    v2f b; b.x = bf.x; b.y = bf.y;
    float2 a0f = aP[(kk >> 1)];
    float2 a1f = aP[(kk >> 1) + rowStride2];
    float2 a2f = aP[(kk >> 1) + 2 * rowStride2];
    v2f a0; a0.x = a0f.x; a0.y = a0f.y;
    v2f a1; a1.x = a1f.x; a1.y = a1f.y;
    v2f a2; a2.x = a2f.x; a2.y = a2f.y;
    c0 = __builtin_amdgcn_wmma_f32_16x16x4_f32(false, a0, false, b, (short)0, c0, false, false);
    c1 = __builtin_amdgcn_wmma_f32_16x16x4_f32(false, a1, false, b, (short)0, c1, false, false);
    c2 = __builtin_amdgcn_wmma_f32_16x16x4_f32(false, a2, false, b, (short)0, c2, false, false);
  }

  // write h (+bias) tiles to LDS, rows [0,48)
  const float bv = be[n0 + laneN];
#pragma unroll
  for (int i = 0; i < 8; ++i) {
    hT[(i + rOff) * LSTRIDE + n0 + laneN]        = c0[i] + bv;
    hT[(16 + i + rOff) * LSTRIDE + n0 + laneN]   = c1[i] + bv;
    hT[(32 + i + rOff) * LSTRIDE + n0 + laneN]   = c2[i] + bv;
  }
  __syncthreads();

  // ---- LayerNorm: wave handles rows {wave, wave+16, wave+32}; lane owns 8 cols ----
  const float4 ga0 = *reinterpret_cast<const float4*>(gamma + 4 * lane);
  const float4 ga1 = *reinterpret_cast<const float4*>(gamma + 128 + 4 * lane);
  const float4 bt0 = *reinterpret_cast<const float4*>(beta + 4 * lane);
  const float4 bt1 = *reinterpret_cast<const float4*>(beta + 128 + 4 * lane);

#pragma unroll
  for (int t = 0; t < 3; ++t) {
    const int row = wave + 16 * t;
    const float4 va = *reinterpret_cast<const float4*>(&hT[row * LSTRIDE + 4 * lane]);
    const float4 vb = *reinterpret_cast<const float4*>(&hT[row * LSTRIDE + 128 + 4 * lane]);

    float s  = va.x + va.y + va.z + va.w + vb.x + vb.y + vb.z + vb.w;
    float s2 = va.x * va.x + va.y * va.y + va.z * va.z + va.w * va.w
             + vb.x * vb.x + vb.y * vb.y + vb.z * vb.z + vb.w * vb.w;
#pragma unroll
    for (int off = 16; off >= 1; off >>= 1) {
      s  += __shfl_xor(s,  off, 32);
      s2 += __shfl_xor(s2, off, 32);
    }
    const float mu  = s  * (1.f / MEM_DIM);
    const float var = s2 * (1.f / MEM_DIM) - mu * mu;
    const float inv = rsqrtf(var + 1e-6f);

    float4 oa, ob;
    oa.x = (va.x - mu) * inv * ga0.x + bt0.x;
    oa.y = (va.y - mu) * inv * ga0.y + bt0.y;
    oa.z = (va.z - mu) * inv * ga0.z + bt0.z;
    oa.w = (va.w - mu) * inv * ga0.w + bt0.w;
    ob.x = (vb.x - mu) * inv * ga1.x + bt1.x;
    ob.y = (vb.y - mu) * inv * ga1.y + bt1.y;
    ob.z = (vb.z - mu) * inv * ga1.z + bt1.z;
    ob.w = (vb.w - mu) * inv * ga1.w + bt1.w;

    const size_t gBase = ((size_t)bso1 * NO + row) * MEM_DIM;
    *reinterpret_cast<float4*>(out + gBase + 4 * lane)       = oa;
    *reinterpret_cast<float4*>(out + gBase + 128 + 4 * lane) = ob;
  }
}

// ---------------------------------------------------------------------------
extern "C" void kernel_launch(void* const* d_in, const int* in_sizes, int n_in,
                              void* d_out, int out_size, void* d_ws, size_t ws_size,
                              hipStream_t stream) {
  const float* rel   = (const float*)d_in[0];
  const float* W1    = (const float*)d_in[1];
  const float* b1    = (const float*)d_in[2];
  const float* W2    = (const float*)d_in[3];
  const float* b2    = (const float*)d_in[4];
  const float* We    = (const float*)d_in[5];
  const float* be    = (const float*)d_in[6];
  const float* gamma = (const float*)d_in[7];
  const float* beta  = (const float*)d_in[8];
  float* out = (float*)d_out;

  float* w   = (float*)d_ws;
  float* p1  = w;                                   // 4608*256
  float* p2  = p1  + (size_t)ROWS_P * MEM_DIM;      // 4608*256
  float* W1T = p2  + (size_t)ROWS_P * MEM_DIM;      // 512*256
  float* W2T = W1T + (size_t)REL_DIM * MEM_DIM;     // 512*256
  float* WeT = W2T + (size_t)REL_DIM * MEM_DIM;     // 256*256

  transpose_kernel<<<dim3((REL_DIM * MEM_DIM) / 256), 256, 0, stream>>>(W1, W1T, REL_DIM, MEM_DIM);
  transpose_kernel<<<dim3((REL_DIM * MEM_DIM) / 256), 256, 0, stream>>>(W2, W2T, REL_DIM, MEM_DIM);
  transpose_kernel<<<dim3((MEM_DIM * MEM_DIM) / 256), 256, 0, stream>>>(We, WeT, MEM_DIM, MEM_DIM);

  proj_kernel<<<dim3(ROWS_P / 16, 2), 512, 0, stream>>>(rel, W1T, b1, W2T, b2, p1, p2);

  const size_t smemBytes = (size_t)(2 * NO) * LSTRIDE * sizeof(float);  // 99,840 B
  encode_kernel<<<dim3(ROWS_P), 512, smemBytes, stream>>>(p1, p2, WeT, be, gamma, beta, out);
}